// NashCascadeNeuralNetwork_45509473468984
// MI455X (gfx1250) — compile-verified
//
#include <hip/hip_runtime.h>
#include <cstddef>

// Problem constants (match reference)
#define LN 3
#define BN 2048
#define SN 2048
#define TN 16
#define G_ACC 9.81f
#define TWO_G (2.0f * 9.81f)
#define NCHUNK (SN / 8)
#define NROWS 512   // 64 waves x 8 group-partials (4 buckets per group)

typedef __attribute__((ext_vector_type(2))) float v2f;
typedef __attribute__((ext_vector_type(8))) float v8f;

// ---------------------------------------------------------------------------
// Solve one layer: one thread (lane) per bucket, one wave per block.
// Software-pipelined chunks of 8 spigots:
//   loads(k+1) -> serial chain(k) -> depth-2 xor-tree(k) -> b128 stores(k)
//   -> precompute cf/Bv(k+1)
// Lanes 0..7 each own one partial row (sum over a fixed 4-bucket group) and
// hold all 8 consecutive columns of a chunk in registers -> 2 b128 stores.
// ---------------------------------------------------------------------------
__global__ __launch_bounds__(32) void nash_solve_kernel(
    const float* __restrict__ theta,     // [BN*SN] this layer
    const float* __restrict__ spig,      // [BN*SN*2] this layer (h,a interleaved)
    const float* __restrict__ u,         // [TN]
    const float* __restrict__ colsum_in, // [SN] col sums of previous layer
    float* __restrict__ H,               // [BN] head state for this layer (ws)
    float* __restrict__ partial,         // [NROWS*SN]
    int t, int layer)
{
    const int lane = (int)threadIdx.x;          // 0..31
    const int b    = (int)blockIdx.x * 32 + lane;

    const float ppl = u[t] * (1.0f / (float)LN);
    float inflow;
    if (layer == 0) {
        inflow = (b == 0) ? ppl : 0.0f;
    } else {
        inflow = colsum_in[b] + ppl * (1.0f / (float)BN);
    }

    const float Hb   = H[b] + inflow;
    const float base = TWO_G * Hb;              // 2G*Hb
    float cum = 0.0f;

    const float4* thv = (const float4*)(theta + (size_t)b * SN);      // 2 per chunk
    const float4* spv = (const float4*)(spig  + (size_t)b * SN * 2);  // 4 per chunk
    float* prowbase = partial + (size_t)((int)blockIdx.x * 8 + (lane & 7)) * SN;

    float cf[8], Bv[8];
    auto precompute = [&](const float4& t0, const float4& t1,
                          const float4& s0, const float4& s1,
                          const float4& s2, const float4& s3) {
        cf[0] = t0.x * s0.y;  Bv[0] = fmaf(-TWO_G, s0.x, base);
        cf[1] = t0.y * s0.w;  Bv[1] = fmaf(-TWO_G, s0.z, base);
        cf[2] = t0.z * s1.y;  Bv[2] = fmaf(-TWO_G, s1.x, base);
        cf[3] = t0.w * s1.w;  Bv[3] = fmaf(-TWO_G, s1.z, base);
        cf[4] = t1.x * s2.y;  Bv[4] = fmaf(-TWO_G, s2.x, base);
        cf[5] = t1.y * s2.w;  Bv[5] = fmaf(-TWO_G, s2.z, base);
        cf[6] = t1.z * s3.y;  Bv[6] = fmaf(-TWO_G, s3.x, base);
        cf[7] = t1.w * s3.w;  Bv[7] = fmaf(-TWO_G, s3.z, base);
    };

    // Prologue: chunk 0
    precompute(thv[0], thv[1], spv[0], spv[1], spv[2], spv[3]);

    for (int k = 0; k < NCHUNK; ++k) {
        // Issue next-chunk loads early (latency hidden behind chain + tree)
        const int kn = (k + 1 < NCHUNK) ? k + 1 : k;
        float4 nt0 = thv[2 * kn],     nt1 = thv[2 * kn + 1];
        float4 ns0 = spv[4 * kn],     ns1 = spv[4 * kn + 1];
        float4 ns2 = spv[4 * kn + 2], ns3 = spv[4 * kn + 3];
        const int kpf = (k + 4 < NCHUNK) ? k + 4 : k;   // L2 prefetch ahead
        __builtin_prefetch(thv + 2 * kpf, 0, 0);
        __builtin_prefetch(spv + 4 * kpf, 0, 0);

        // Serial recurrence: fma -> max -> v_sqrt -> fma per spigot
        float q[8];
#pragma unroll
        for (int j = 0; j < 8; ++j) {
            float y = fmaf(-G_ACC, cum, Bv[j]);   // 2G*(Hb - h - cum/2)
            y = fmaxf(y, 0.0f);
#if __has_builtin(__builtin_amdgcn_sqrtf)
            float sq = __builtin_amdgcn_sqrtf(y); // v_sqrt_f32
#else
            float sq = sqrtf(y);
#endif
            float qq = cf[j] * sq;
            cum += qq;
            q[j] = qq;
        }

        // Depth-2 deterministic tree: 8 independent swizzles per level
#pragma unroll
        for (int j = 0; j < 8; ++j) q[j] += __shfl_xor(q[j], 16, 32);
#pragma unroll
        for (int j = 0; j < 8; ++j) q[j] += __shfl_xor(q[j], 8, 32);

        // Lane L<8 now holds group-L (buckets {L, L+8, L+16, L+24}) sums for
        // columns k*8 .. k*8+7 -> one partial row, two b128 stores.
        if (lane < 8) {
            float4* dst = (float4*)(prowbase + k * 8);
            dst[0] = make_float4(q[0], q[1], q[2], q[3]);
            dst[1] = make_float4(q[4], q[5], q[6], q[7]);
        }

        // Precompute next chunk's cf/Bv (independent of tree results)
        precompute(nt0, nt1, ns0, ns1, ns2, ns3);
    }

    H[b] = Hb - cum;
}

// ---------------------------------------------------------------------------
// Column reduction of partial[NROWS][SN] -> colsum[SN] using f32 WMMA:
// colsum = ones(16x4) x P(4x16) chained over K. One wave per 16-column tile.
// Fully deterministic, full f32 precision.
// ---------------------------------------------------------------------------
__global__ __launch_bounds__(128) void nash_reduce_kernel(
    const float* __restrict__ P,       // [NROWS*SN]
    float* __restrict__ colsum)        // [SN]
{
    const int wave = ((int)blockIdx.x * (int)blockDim.x + (int)threadIdx.x) >> 5;
    const int lane = (int)threadIdx.x & 31;
    const int c0   = wave * 16;                 // column tile base
    const int n    = lane & 15;                 // column within tile
    const int kr   = (lane >> 4) << 1;          // 0 (lanes 0-15) or 2 (16-31)

#if __has_builtin(__builtin_amdgcn_wmma_f32_16x16x4_f32)
    v8f c = {};
    v2f a; a.x = 1.0f; a.y = 1.0f;              // all-ones A (layout-invariant)
    for (int kb = 0; kb < NROWS; kb += 4) {
        // B(4x16): VGPR0 = row kb+kr, VGPR1 = row kb+kr+1, per lane-half
        v2f bv;
        bv.x = P[(size_t)(kb + kr)     * SN + c0 + n];
        bv.y = P[(size_t)(kb + kr + 1) * SN + c0 + n];
        c = __builtin_amdgcn_wmma_f32_16x16x4_f32(
                false, a, false, bv, (short)0, c, false, false);
    }
    // D row 0: lanes 0-15 hold colsum for columns c0..c0+15
    if (lane < 16) colsum[c0 + lane] = c[0];
#else
    if (lane < 16) {
        float s = 0.0f;
        for (int r = 0; r < NROWS; ++r) s += P[(size_t)r * SN + c0 + lane];
        colsum[c0 + lane] = s;
    }
#endif
}

// ---------------------------------------------------------------------------
// Final deterministic tree sum of colsum[SN] -> outflows[t]
// ---------------------------------------------------------------------------
__global__ __launch_bounds__(256) void nash_final_kernel(
    const float* __restrict__ colsum, float* __restrict__ out)
{
    __shared__ float sm[256];
    const int tid = (int)threadIdx.x;
    float s = 0.0f;
    for (int i = tid; i < SN; i += 256) s += colsum[i];
    sm[tid] = s;
    __syncthreads();
    for (int off = 128; off > 0; off >>= 1) {
        if (tid < off) sm[tid] += sm[tid + off];
        __syncthreads();
    }
    if (tid == 0) *out = sm[0];
}

// ---------------------------------------------------------------------------
// Host driver: T=16 timesteps x L=3 layers, all on `stream` (graph-safe).
// ws layout: [0,24KB) H | [32KB,40KB) colsum | [64KB, 64KB+4MB) partial
// ---------------------------------------------------------------------------
extern "C" void kernel_launch(void* const* d_in, const int* in_sizes, int n_in,
                              void* d_out, int out_size, void* d_ws, size_t ws_size,
                              hipStream_t stream)
{
    const float* u     = (const float*)d_in[0];   // [TN]
    const float* theta = (const float*)d_in[1];   // [LN*BN*SN]
    const float* spig  = (const float*)d_in[2];   // [LN*BN*SN*2]
    const float* H0    = (const float*)d_in[3];   // [LN*BN]
    float* out = (float*)d_out;                   // [TN]

    char* ws = (char*)d_ws;
    float* H       = (float*)(ws);                // LN*BN floats
    float* colsum  = (float*)(ws + 32768);        // SN floats
    float* partial = (float*)(ws + 65536);        // NROWS*SN floats (4 MB)

    // Re-initialize state from inputs every call (deterministic, no input mutation)
    hipMemcpyAsync(H, H0, (size_t)LN * BN * sizeof(float),
                   hipMemcpyDeviceToDevice, stream);

    for (int t = 0; t < TN; ++t) {
        for (int l = 0; l < LN; ++l) {
            nash_solve_kernel<<<BN / 32, 32, 0, stream>>>(
                theta + (size_t)l * BN * SN,
                spig  + (size_t)l * BN * SN * 2,
                u, colsum, H + (size_t)l * BN, partial, t, l);
            nash_reduce_kernel<<<(SN / 16) / 4, 128, 0, stream>>>(partial, colsum);
        }
        nash_final_kernel<<<1, 256, 0, stream>>>(colsum, out + t);
    }
}